// CompressibleOgden_86036784874065
// MI455X (gfx1250) — compile-verified
//
#include <hip/hip_runtime.h>
#include <hip/hip_bf16.h>

// Compressible Ogden strain energy, N points, each with a 3x3 deformation
// gradient F (row-major, AoS: point p at F + 9p).
//
// MI455X strategy: bandwidth/VALU-balanced elementwise kernel. No matmul
// structure -> no WMMA (a per-point 3x3 F^T F is 27 FMAs; WMMA marshaling
// would cost more than it saves). The CDNA5 win is the async global->LDS DMA
// pipeline (ASYNCcnt): stage 36-byte AoS records through LDS (odd stride 9 =>
// conflict-free banks), double-buffered so the next chunk's DMA overlaps this
// chunk's transcendental-heavy compute. Default RT caching: the 36 MB input
// fits in the 192 MB L2, so graph replays run at L2 bandwidth.

#define BLK        256          // threads per block (8 wave32)
#define PTS        256          // points per chunk
#define FPC        (PTS * 9)    // floats per chunk = 2304
#define CHUNK_B    (FPC * 4)    // bytes per chunk  = 9216

// Issue exactly 3 async-to-LDS ops per thread (2 x b128 covering dwords
// [0,2048), 1 x b32 covering dwords [2048,2304)). All lanes always active so
// every wave's ASYNCcnt increments by exactly 3 per stage.
__device__ __forceinline__ void stage_chunk(const float* __restrict__ F,
                                            unsigned ldsBase,
                                            long long floatBase,
                                            long long totalFloats,
                                            int tid) {
  const long long maxF4 = totalFloats >> 2;   // # of full float4's in F
  const long long f4base = floatBase >> 2;    // chunk base is 2304*chunk (div by 4)
#pragma unroll
  for (int r = 0; r < 2; ++r) {
    long long gi4 = f4base + (long long)(r * BLK + tid);
    if (gi4 > maxF4 - 1) gi4 = maxF4 - 1;     // tail clamp (dup loads, ignored later)
    unsigned ldsAddr = ldsBase + (unsigned)(r * BLK + tid) * 16u;
    unsigned gOff = (unsigned)(gi4 << 4);     // byte offset from SADDR base
    asm volatile("global_load_async_to_lds_b128 %0, %1, %2"
                 :
                 : "v"(ldsAddr), "v"(gOff), "s"(F)
                 : "memory");
  }
  long long gd = floatBase + 2048 + tid;      // remaining 256 dwords
  if (gd > totalFloats - 1) gd = totalFloats - 1;
  unsigned ldsAddr = ldsBase + 8192u + (unsigned)tid * 4u;
  unsigned gOff = (unsigned)(gd << 2);
  asm volatile("global_load_async_to_lds_b32 %0, %1, %2"
               :
               : "v"(ldsAddr), "v"(gOff), "s"(F)
               : "memory");
}

__global__ __launch_bounds__(BLK) void CompressibleOgden_86036784874065_kernel(
    const float* __restrict__ F, const float* __restrict__ mu,
    const float* __restrict__ alpha, float* __restrict__ out, int n) {
  __shared__ float lds[2 * FPC];              // double buffer, 18432 B

  const int tid = threadIdx.x;
  const long long totalFloats = 9LL * (long long)n;
  const int nChunks = (n + PTS - 1) / PTS;

  int chunk = (int)blockIdx.x;
  if (chunk >= nChunks) return;

  // Uniform material parameters (tiny, uniform loads, hoisted out of loop).
  const float w0 = mu[0] / alpha[0], w1 = mu[1] / alpha[1], w2 = mu[2] / alpha[2];
  const float h0 = 0.5f * alpha[0], h1 = 0.5f * alpha[1], h2 = 0.5f * alpha[2];

  const unsigned buf0 = (unsigned)(size_t)&lds[0];
  const unsigned buf1 = (unsigned)(size_t)&lds[FPC];

  stage_chunk(F, buf0, (long long)chunk * FPC, totalFloats, tid);
  int cur = 0;

  for (;;) {
    const int next = chunk + (int)gridDim.x;
    const bool hasNext = (next < nChunks);
    if (hasNext) {
      // Prefetch next chunk into the other buffer while we compute this one.
      stage_chunk(F, cur ? buf0 : buf1, (long long)next * FPC, totalFloats, tid);
      asm volatile("s_wait_asynccnt 0x3" ::: "memory");  // current chunk landed
    } else {
      asm volatile("s_wait_asynccnt 0x0" ::: "memory");
    }
    __syncthreads();

    // ---- per-point math (fp32) ----
    const float* fp = &lds[(cur ? FPC : 0) + tid * 9];   // stride 9: no bank conflicts
    const float f00 = fp[0], f01 = fp[1], f02 = fp[2];
    const float f10 = fp[3], f11 = fp[4], f12 = fp[5];
    const float f20 = fp[6], f21 = fp[7], f22 = fp[8];

    // C = F^T F (symmetric)
    const float c00 = f00 * f00 + f10 * f10 + f20 * f20;
    const float c11 = f01 * f01 + f11 * f11 + f21 * f21;
    const float c22 = f02 * f02 + f12 * f12 + f22 * f22;
    const float c01 = f00 * f01 + f10 * f11 + f20 * f21;
    const float c02 = f00 * f02 + f10 * f12 + f20 * f22;
    const float c12 = f01 * f02 + f11 * f12 + f21 * f22;

    const float detC = c00 * (c11 * c22 - c12 * c12)
                     - c01 * (c01 * c22 - c12 * c02)
                     + c02 * (c01 * c12 - c11 * c02);      // = J^2

    // Closed-form symmetric 3x3 eigenvalues (trigonometric method)
    const float q  = (c00 + c11 + c22) * (1.0f / 3.0f);
    const float b00 = c00 - q, b11 = c11 - q, b22 = c22 - q;
    const float p1 = c01 * c01 + c02 * c02 + c12 * c12;
    const float p2 = b00 * b00 + b11 * b11 + b22 * b22 + 2.0f * p1;
    const float p  = __builtin_amdgcn_sqrtf(p2 * (1.0f / 6.0f) + 1e-30f);
    const float invp = __builtin_amdgcn_rcpf(p);
    const float detS = b00 * (b11 * b22 - c12 * c12)
                     - c01 * (c01 * b22 - c12 * c02)
                     + c02 * (c01 * c12 - b11 * c02);
    float r = 0.5f * detS * invp * invp * invp;
    r = fminf(1.0f, fmaxf(-1.0f, r));
    const float phi = acosf(r) * (1.0f / 3.0f);
    float sp, cp;
    __sincosf(phi, &sp, &cp);
    const float twop = 2.0f * p;
    const float e1 = q + twop * cp;
    const float e3 = q + twop * (-0.5f * cp - 0.86602540378443865f * sp);
    const float e2 = 3.0f * q - e1 - e3;

    // lambda_bar_i = e_i * detC^(-1/3) (J^(-2/3) scaling of eigenvalues)
    // => log2(lambda_bar_i) = log2(e_i) - (1/3) log2(detC)   (saves an exp2)
    const float l2d = __log2f(detC);
    const float s3  = l2d * (1.0f / 3.0f);
    const float t1 = __log2f(e1) - s3;
    const float t2 = __log2f(e2) - s3;
    const float t3 = __log2f(e3) - s3;

    // pw_k = sum_i lambda_bar_i^(alpha_k/2) = sum_i exp2(h_k * t_i)
    const float pw0 = __builtin_amdgcn_exp2f(h0 * t1)
                    + __builtin_amdgcn_exp2f(h0 * t2)
                    + __builtin_amdgcn_exp2f(h0 * t3);
    const float pw1 = __builtin_amdgcn_exp2f(h1 * t1)
                    + __builtin_amdgcn_exp2f(h1 * t2)
                    + __builtin_amdgcn_exp2f(h1 * t3);
    const float pw2 = __builtin_amdgcn_exp2f(h2 * t1)
                    + __builtin_amdgcn_exp2f(h2 * t2)
                    + __builtin_amdgcn_exp2f(h2 * t3);

    const float W_iso = w0 * (pw0 - 3.0f) + w1 * (pw1 - 3.0f) + w2 * (pw2 - 3.0f);
    // KAPPA=100, BETA=2: W_vol = 25*(J^2 - 2*ln J - 1) = 25*(detC - ln detC - 1)
    const float W_vol = 25.0f * (detC - l2d * 0.69314718055994531f - 1.0f);

    const long long pt = (long long)chunk * PTS + tid;
    if (pt < n) out[pt] = W_iso + W_vol;

    if (!hasNext) break;
    __syncthreads();   // everyone done reading 'cur' before it is re-staged
    chunk = next;
    cur ^= 1;
  }
}

extern "C" void kernel_launch(void* const* d_in, const int* in_sizes, int n_in,
                              void* d_out, int out_size, void* d_ws, size_t ws_size,
                              hipStream_t stream) {
  const float* F     = (const float*)d_in[0];
  const float* mu    = (const float*)d_in[1];
  const float* alpha = (const float*)d_in[2];
  float* out = (float*)d_out;
  const int n = in_sizes[0] / 9;
  const int nChunks = (n + PTS - 1) / PTS;
  int grid = nChunks < 1024 ? nChunks : 1024;
  if (grid < 1) grid = 1;
  CompressibleOgden_86036784874065_kernel<<<dim3(grid), dim3(BLK), 0, stream>>>(
      F, mu, alpha, out, n);
}